// GraphAttentionLayer_9612136808772
// MI455X (gfx1250) — compile-verified
//
#include <hip/hip_runtime.h>
#include <hip/hip_bf16.h>

// ---------------------------------------------------------------------------
// GAT layer for MI455X (gfx1250, wave32).
// Phase 1: Wh = h @ W^T + b  via V_WMMA_F32_16X16X4_F32 (exact f32 WMMA).
// Phase 2: per-node alpha_src/alpha_dst, init m/-inf, denom/0, zero out.
// Phase 3/4/5: segment-softmax over edges + scatter aggregation (L2-resident).
// ---------------------------------------------------------------------------

typedef __attribute__((ext_vector_type(2))) float v2f;
typedef __attribute__((ext_vector_type(8))) float v8f;

static constexpr int N_NODES  = 50000;
static constexpr int N_EDGES  = 800000;
static constexpr int DIM      = 256;    // IN_DIM == OUT_DIM
static constexpr int HEADS    = 4;
static constexpr int HEAD_DIM = 64;
static constexpr float NEG_SLOPE = 0.2f;

// LDS row stride for the staged h slab: 260 floats.
// A-fragment b64 read banks = (4*row + k) mod 64 -> all 64 banks hit once.
static constexpr int HS_STRIDE = 260;

// ---------------------------------------------------------------------------
// Kernel 1: Wh = h @ W^T + b.  One block = 16 output rows x 256 output cols.
// 8 waves; each wave owns two 16x16 column tiles and shares one A fragment.
// ---------------------------------------------------------------------------
__global__ __launch_bounds__(256) void gat_gemm_wmma(
    const float* __restrict__ h, const float* __restrict__ W,
    const float* __restrict__ bias, float* __restrict__ Wh) {
  __shared__ float hs[16 * HS_STRIDE];

  const int row0 = blockIdx.x * 16;
  const int t    = threadIdx.x;

  // Stage 16x256 slab of h into LDS (coalesced b32 per thread).
  #pragma unroll
  for (int r = 0; r < 16; ++r)
    hs[r * HS_STRIDE + t] = h[(row0 + r) * DIM + t];
  __syncthreads();

  const int wave = t >> 5;          // 0..7
  const int lane = t & 31;
  const int half = lane >> 4;       // 0/1 : selects K sub-pair per ISA layout
  const int lr   = lane & 15;       // row (A) / col (B,C,D) index

  const int col0 = wave * 16 + lr;          // tile 0 global column
  const int col1 = (wave + 8) * 16 + lr;    // tile 1 global column
  const float* __restrict__ w0 = W + col0 * DIM;
  const float* __restrict__ w1 = W + col1 * DIM;

  v8f c0 = {};
  v8f c1 = {};

  for (int k0 = 0; k0 < DIM; k0 += 4) {
    const int k = k0 + 2 * half;
    // A fragment: A[lr][k], A[lr][k+1]  (16x4 f32 layout, 2 VGPRs)
    v2f a = *(const v2f*)&hs[lr * HS_STRIDE + k];
    // B fragment: B[k][col] = W[col][k] -> contiguous pair in W's row
    v2f b0 = *(const v2f*)(w0 + k);
    v2f b1 = *(const v2f*)(w1 + k);
    // D = A x B + C   (8 args: neg_a, A, neg_b, B, c_mod, C, reuse_a, reuse_b)
    c0 = __builtin_amdgcn_wmma_f32_16x16x4_f32(false, a, false, b0,
                                               (short)0, c0, false, false);
    c1 = __builtin_amdgcn_wmma_f32_16x16x4_f32(false, a, false, b1,
                                               (short)0, c1, false, false);
  }

  // Epilogue: D vgpr r holds row (r + 8*half), col = lr. Add bias, store.
  const float bias0 = bias[col0];
  const float bias1 = bias[col1];
  #pragma unroll
  for (int r = 0; r < 8; ++r) {
    const int row = row0 + r + 8 * half;
    Wh[row * DIM + col0] = c0[r] + bias0;
    Wh[row * DIM + col1] = c1[r] + bias1;
  }
}

// ---------------------------------------------------------------------------
// Kernel 2: per-node alpha_src / alpha_dst; init m=-inf, denom=0, out=0.
// ---------------------------------------------------------------------------
__global__ __launch_bounds__(256) void node_prep(
    const float* __restrict__ Wh, const float* __restrict__ a_src,
    const float* __restrict__ a_dst, float* __restrict__ alpha_src,
    float* __restrict__ alpha_dst, float* __restrict__ m,
    float* __restrict__ denom, float* __restrict__ out) {
  __shared__ float ss[256];
  __shared__ float sd[256];
  const int n = blockIdx.x;
  const int t = threadIdx.x;

  const float w = Wh[n * DIM + t];
  out[n * DIM + t] = 0.0f;          // harness poisons d_out; we scatter-add later
  ss[t] = w * a_src[t];             // a_src flat [H*HEAD_DIM] matches t
  sd[t] = w * a_dst[t];
  __syncthreads();

  if (t < HEADS) {
    float s1 = 0.0f, s2 = 0.0f;
    #pragma unroll
    for (int d = 0; d < HEAD_DIM; ++d) {
      s1 += ss[t * HEAD_DIM + d];
      s2 += sd[t * HEAD_DIM + d];
    }
    alpha_src[n * HEADS + t] = s1;
    alpha_dst[n * HEADS + t] = s2;
    m[n * HEADS + t]     = __int_as_float(0xFF800000); // -inf
    denom[n * HEADS + t] = 0.0f;
  }
}

// Ordered-float atomic max: signed-int max for non-negatives,
// unsigned-int min for negatives. Initial value must be -inf.
__device__ __forceinline__ void atomic_max_f32(float* addr, float val) {
  const int vi = __float_as_int(val);
  if (vi >= 0)
    atomicMax((int*)addr, vi);
  else
    atomicMin((unsigned int*)addr, (unsigned int)vi);
}

__device__ __forceinline__ float leaky(float x) {
  return x > 0.0f ? x : NEG_SLOPE * x;
}

// ---------------------------------------------------------------------------
// Kernel 3: segment max of edge logits into m[i,h].
// ---------------------------------------------------------------------------
__global__ __launch_bounds__(256) void edge_max(
    const int* __restrict__ ei, const float* __restrict__ alpha_src,
    const float* __restrict__ alpha_dst, float* __restrict__ m) {
  const int idx = blockIdx.x * 256 + threadIdx.x;
  if (idx >= N_EDGES * HEADS) return;
  const int e  = idx >> 2;
  const int hh = idx & 3;
  const int i = ei[e];
  const int j = ei[N_EDGES + e];
  const float v = leaky(alpha_src[i * HEADS + hh] + alpha_dst[j * HEADS + hh]);
  atomic_max_f32(&m[i * HEADS + hh], v);
}

// ---------------------------------------------------------------------------
// Kernel 4: denom[i,h] += exp(e - m[i,h]).
// ---------------------------------------------------------------------------
__global__ __launch_bounds__(256) void edge_sum(
    const int* __restrict__ ei, const float* __restrict__ alpha_src,
    const float* __restrict__ alpha_dst, const float* __restrict__ m,
    float* __restrict__ denom) {
  const int idx = blockIdx.x * 256 + threadIdx.x;
  if (idx >= N_EDGES * HEADS) return;
  const int e  = idx >> 2;
  const int hh = idx & 3;
  const int i = ei[e];
  const int j = ei[N_EDGES + e];
  const float v = leaky(alpha_src[i * HEADS + hh] + alpha_dst[j * HEADS + hh]);
  atomicAdd(&denom[i * HEADS + hh], __expf(v - m[i * HEADS + hh]));
}

// ---------------------------------------------------------------------------
// Kernel 5: out[i,:] += att * Wh[j,:]. One 256-thread block per edge:
// coalesced gather of Wh row j, coalesced atomic scatter into out row i.
// ---------------------------------------------------------------------------
__global__ __launch_bounds__(256) void edge_agg(
    const int* __restrict__ ei, const float* __restrict__ alpha_src,
    const float* __restrict__ alpha_dst, const float* __restrict__ m,
    const float* __restrict__ denom, const float* __restrict__ Wh,
    float* __restrict__ out) {
  const int e  = blockIdx.x;
  const int t  = threadIdx.x;
  const int hh = t >> 6;            // head of this column
  const int i = ei[e];
  const int j = ei[N_EDGES + e];
  const float v   = leaky(alpha_src[i * HEADS + hh] + alpha_dst[j * HEADS + hh]);
  const float att = __expf(v - m[i * HEADS + hh]) / denom[i * HEADS + hh];
  atomicAdd(&out[i * DIM + t], att * Wh[j * DIM + t]);
}

// ---------------------------------------------------------------------------
// Launch wrapper.
// Inputs (setup_inputs order): h [N,256] f32, edge_index [2,E] i32,
//   W [256,256] f32, b [256] f32, a_src [4,64] f32, a_dst [4,64] f32.
// Output: [N,256] f32.
// Workspace: Wh (N*256) | alpha_src (N*4) | alpha_dst (N*4) | m (N*4) | denom (N*4)
// ---------------------------------------------------------------------------
extern "C" void kernel_launch(void* const* d_in, const int* in_sizes, int n_in,
                              void* d_out, int out_size, void* d_ws, size_t ws_size,
                              hipStream_t stream) {
  const float* h     = (const float*)d_in[0];
  const int*   ei    = (const int*)d_in[1];
  const float* W     = (const float*)d_in[2];
  const float* bias  = (const float*)d_in[3];
  const float* a_src = (const float*)d_in[4];
  const float* a_dst = (const float*)d_in[5];
  float* out = (float*)d_out;

  float* Wh        = (float*)d_ws;
  float* alpha_src = Wh + (size_t)N_NODES * DIM;
  float* alpha_dst = alpha_src + (size_t)N_NODES * HEADS;
  float* m         = alpha_dst + (size_t)N_NODES * HEADS;
  float* denom     = m + (size_t)N_NODES * HEADS;

  // Phase 1: dense GEMM via WMMA. N=50000 -> exactly 3125 16-row slabs.
  gat_gemm_wmma<<<N_NODES / 16, 256, 0, stream>>>(h, W, bias, Wh);

  // Phase 2: per-node logit components + init.
  node_prep<<<N_NODES, 256, 0, stream>>>(Wh, a_src, a_dst, alpha_src, alpha_dst,
                                         m, denom, out);

  // Phase 3/4: segment softmax statistics over edges.
  const int eh_blocks = (N_EDGES * HEADS + 255) / 256;
  edge_max<<<eh_blocks, 256, 0, stream>>>(ei, alpha_src, alpha_dst, m);
  edge_sum<<<eh_blocks, 256, 0, stream>>>(ei, alpha_src, alpha_dst, m, denom);

  // Phase 5: weighted scatter aggregation.
  edge_agg<<<N_EDGES, 256, 0, stream>>>(ei, alpha_src, alpha_dst, m, denom, Wh,
                                        out);
}